// BilinearClassification_85117661872526
// MI455X (gfx1250) — compile-verified
//
#include <hip/hip_runtime.h>

// Shapes from the reference
#define B_  32
#define L_  512
#define H_  768
#define E_  64
#define S_  8
#define T_  128
#define NT_ 42
#define NP_ 48         // NT padded to 3 full 16-wide N tiles
#define F_  (H_ * 8)   // 6144 bilinear features

#define CC  16         // c-chunk (each c covers 8 H values -> 64 f values)
#define CH  (CC * 8)   // 128 floats per row per chunk
#define LDP 132        // padded LDS row stride (16B aligned, conflict-free)

typedef __attribute__((ext_vector_type(2))) float v2f;
typedef __attribute__((ext_vector_type(8))) float v8f;

// ---------------------------------------------------------------------------
// Kernel 1: masked mean over ragged subword spans -> ent[B,E,H] in workspace
// ---------------------------------------------------------------------------
__global__ __launch_bounds__(192) void entity_mean_kernel(
    const float* __restrict__ hidden,   // [B,L,H]
    const int*   __restrict__ subw_idx, // [B,E,S]
    const int*   __restrict__ subw_msk, // [B,E,S] (0/1)
    float*       __restrict__ ent)      // [B,E,H]
{
  const int be    = blockIdx.x;     // b*E + e
  const int b     = be >> 6;        // / E_
  const int h0    = threadIdx.x * 4;
  const int ibase = be * S_;

  float4 acc = make_float4(0.f, 0.f, 0.f, 0.f);
  int cnt = 0;
  #pragma unroll
  for (int s = 0; s < S_; ++s) {
    if (subw_msk[ibase + s]) {
      const int row = subw_idx[ibase + s];
      const float4 v =
          *(const float4*)(hidden + ((size_t)b * L_ + row) * H_ + h0);
      acc.x += v.x; acc.y += v.y; acc.z += v.z; acc.w += v.w;
      ++cnt;
    }
  }
  const float inv = 1.0f / (float)(cnt > 0 ? cnt : 1);
  float4 r;
  r.x = acc.x * inv; r.y = acc.y * inv; r.z = acc.z * inv; r.w = acc.w * inv;
  *(float4*)(ent + (size_t)be * H_ + h0) = r;
}

// ---------------------------------------------------------------------------
// Kernel 1b: transpose + pad W_fc [F,NT] -> Wt [NP,F] (rows 42..47 zero).
// One-shot, 1.18 MB; makes the GEMM B-operand a contiguous per-lane stream.
// ---------------------------------------------------------------------------
__global__ __launch_bounds__(256) void w_transpose_kernel(
    const float* __restrict__ W,   // [F, NT]
    float*       __restrict__ Wt)  // [NP, F]
{
  const int idx = blockIdx.x * 256 + threadIdx.x;   // over NP*F
  if (idx >= NP_ * F_) return;
  const int n = idx / F_;
  const int f = idx - n * F_;
  Wt[idx] = (n < NT_) ? W[(size_t)f * NT_ + n] : 0.0f;
}

// ---------------------------------------------------------------------------
// Kernel 2: fused (gather subj/obj -> outer-product features -> GEMM -> +bias)
// Grid: 256 blocks of 16 triplet-rows. 3 waves/block, one 16-wide N-tile each.
// D[m,n] accumulated with V_WMMA_F32_16X16X4_F32 over K = 6144.
// A operand (bilinear feature) computed on the fly from LDS subj/obj chunks;
// B operand streamed from the transposed N-major Wt (one b64 load per WMMA).
// ---------------------------------------------------------------------------
__global__ __launch_bounds__(96) void bilinear_wmma_kernel(
    const float* __restrict__ ent,   // [B,E,H] (from kernel 1)
    const int*   __restrict__ trip,  // [B,T,2]
    const int*   __restrict__ pmask, // [B,T]
    const float* __restrict__ Wt,    // [NP, F] transposed weights
    const float* __restrict__ bias,  // [NT]
    float*       __restrict__ out)   // [B,T,NT]
{
  __shared__ float  sS[16][LDP];
  __shared__ float  sO[16][LDP];
  __shared__ size_t baseS[16];
  __shared__ size_t baseO[16];
  __shared__ float  pm16[16];

  const int tid  = threadIdx.x;
  const int lane = tid & 31;
  const int wave = tid >> 5;        // 0..2 -> N tile
  const int tile = blockIdx.x;      // 0..255 -> 16 rows of B*T

  if (tid < 16) {
    const int g  = tile * 16 + tid;
    const int b  = g >> 7;          // / T_
    const int t  = g & (T_ - 1);
    const int si = trip[(b * T_ + t) * 2 + 0];
    const int oi = trip[(b * T_ + t) * 2 + 1];
    baseS[tid] = ((size_t)b * E_ + si) * H_;
    baseO[tid] = ((size_t)b * E_ + oi) * H_;
    pm16[tid]  = pmask[b * T_ + t] ? 1.0f : 0.0f;
  }
  __syncthreads();

  const int   m   = lane & 15;            // output row within tile
  const int   khi = lane >> 4;            // K half: lanes 16-31 hold K+2,K+3
  const int   n   = wave * 16 + (lane & 15);
  const float bn  = bias[(n < NT_) ? n : (NT_ - 1)];
  const float* Wn = Wt + (size_t)n * F_ + (khi * 2);  // lane's weight stream

  v8f acc = {0.f, 0.f, 0.f, 0.f, 0.f, 0.f, 0.f, 0.f};

  for (int c0 = 0; c0 < 96; c0 += CC) {
    __syncthreads();
    // Cooperative stage of this chunk: 2 arrays * 16 rows * 32 float4
    for (int q4 = tid; q4 < 2 * 16 * (CH / 4); q4 += 96) {
      const int which = q4 >> 9;          // 0 = subj, 1 = obj
      const int q     = q4 & 511;
      const int r     = q >> 5;
      const int c4    = q & 31;
      const size_t src =
          (which ? baseO[r] : baseS[r]) + (size_t)c0 * 8 + (size_t)c4 * 4;
      float4 v = *(const float4*)(ent + src);
      const float p = pm16[r];
      float* dst = which ? &sO[r][c4 * 4] : &sS[r][c4 * 4];
      *(float4*)dst = make_float4(v.x * p, v.y * p, v.z * p, v.w * p);
    }
    __syncthreads();

    const int fb0 = c0 * 64;              // feature base of this chunk
    for (int cl = 0; cl < CC; ++cl) {
      const float* rs = &sS[m][cl * 8];
      const float* ro = &sO[m][cl * 8];
      const float* wp = Wn + fb0 + cl * 64;
      // Stream-prefetch the lane's weight row one c-iteration ahead.
      __builtin_prefetch(wp + 64, 0, 1);
      #pragma unroll
      for (int ko = 0; ko < 16; ++ko) {
        const int f0 = ko * 4 + khi * 2;  // this lane's first K elem within c
        v2f A;
        A[0] = rs[(f0)     >> 3] * ro[(f0)     & 7];
        A[1] = rs[(f0 + 1) >> 3] * ro[(f0 + 1) & 7];
        const v2f Bv = *(const v2f*)(wp + ko * 4);  // contiguous, 8B aligned
        acc = __builtin_amdgcn_wmma_f32_16x16x4_f32(
            /*neg_a=*/false, A, /*neg_b=*/false, Bv,
            /*c_mod=*/(short)0, acc, /*reuse_a=*/false, /*reuse_b=*/false);
      }
    }
  }

  if (n < NT_) {
    #pragma unroll
    for (int r = 0; r < 8; ++r) {
      const int mm = r + khi * 8;
      const int g  = tile * 16 + mm;
      out[(size_t)g * NT_ + n] = acc[r] + bn;
    }
  }
}

// ---------------------------------------------------------------------------
extern "C" void kernel_launch(void* const* d_in, const int* in_sizes, int n_in,
                              void* d_out, int out_size, void* d_ws,
                              size_t ws_size, hipStream_t stream) {
  const float* hidden = (const float*)d_in[0];  // [B,L,H]
  const int*   sidx   = (const int*)d_in[1];    // [B,E,S]
  const int*   smask  = (const int*)d_in[2];    // [B,E,S]
  const int*   trip   = (const int*)d_in[3];    // [B,T,2]
  const int*   pmask  = (const int*)d_in[4];    // [B,T]
  const float* W      = (const float*)d_in[5];  // [F,NT]
  const float* bias   = (const float*)d_in[6];  // [NT]
  float*       out    = (float*)d_out;          // [B,T,NT]

  float* ent = (float*)d_ws;                    // [B,E,H]  = 6.0 MB
  float* Wt  = ent + (size_t)B_ * E_ * H_;      // [NP,F]   = 1.18 MB

  entity_mean_kernel<<<B_ * E_, 192, 0, stream>>>(hidden, sidx, smask, ent);
  w_transpose_kernel<<<(NP_ * F_ + 255) / 256, 256, 0, stream>>>(W, Wt);
  bilinear_wmma_kernel<<<(B_ * T_) / 16, 96, 0, stream>>>(ent, trip, pmask, Wt,
                                                          bias, out);
}